// HarmonicElementwiseProduct_26628797235373
// MI455X (gfx1250) — compile-verified
//
#include <hip/hip_runtime.h>
#include <hip/hip_bf16.h>

typedef __attribute__((ext_vector_type(2))) float v2f;
typedef __attribute__((ext_vector_type(8))) float v8f;

#define NCOL  1216   // output row width: 256 (l3=0 block) + 960 (l3=2 block)
#define L3OFF 256

// Preload the combined CG "B matrix" (K x 16, only 6 columns live) into
// per-lane registers following the f32 B layout for 16x16x4:
// lanes 0-15 hold K=4q+{0,1} (VGPR v), lanes 16-31 hold K=4q+{2,3}.
template<int L, int NCH>
__device__ __forceinline__ void load_B(const float* __restrict__ cg0,
                                       const float* __restrict__ cg2,
                                       bool hi, int n, float* Breg)
{
  const int D = 2 * L + 1, K = D * D;
  for (int q = 0; q < NCH; ++q) {
    for (int v = 0; v < 2; ++v) {
      int kk = 4 * q + (hi ? 2 : 0) + v;
      float val = 0.0f;
      if (kk < K && n <= 5) {
        int i = kk / D, j = kk - i * D;
        val = (n == 0) ? cg0[i * D + j]                  // l3=0 column
                       : cg2[(i * D + j) * 5 + (n - 1)]; // l3=2 columns 1..5
      }
      Breg[2 * q + v] = val;
    }
  }
}

// One CG path (l >= 1): build A = outer-product chunks on the fly, accumulate
// D = A x B_cg with V_WMMA_F32_16X16X4_F32, then store the 6 live columns.
// bb is constant across the tile (row0 % 16 == 0), so all 8 stores of a lane
// are a 32-bit offset + constant per-lane stride: straight-line pointer bump
// under a single exec guard.
template<int L, int NCH>
__device__ __forceinline__ void do_path(const float (&r1)[16], const float (&r2)[16],
                                        const float* Breg, bool hi, int n,
                                        unsigned rowBaseOff,  // bb * NCOL
                                        int u0,               // (row0&63) + (hi?8:0)
                                        float* __restrict__ out)
{
  constexpr int D = 2 * L + 1, K = D * D, O = L * L;
  v8f acc = {0.f, 0.f, 0.f, 0.f, 0.f, 0.f, 0.f, 0.f};
#pragma unroll
  for (int q = 0; q < NCH; ++q) {
    v2f A;
#pragma unroll
    for (int v = 0; v < 2; ++v) {
      const int kk0 = 4 * q + v;       // K index if lane in low half
      const int kk1 = 4 * q + 2 + v;   // K index if lane in high half
      const int i0 = (kk0 < K) ? kk0 / D : 0;
      const int j0 = (kk0 < K) ? kk0 % D : 0;
      const int i1 = (kk1 < K) ? kk1 / D : 0;
      const int j1 = (kk1 < K) ? kk1 % D : 0;
      const float p0 = (kk0 < K) ? r1[O + i0] * r2[O + j0] : 0.0f;
      const float p1 = (kk1 < K) ? r1[O + i1] * r2[O + j1] : 0.0f;
      A[v] = hi ? p1 : p0;
    }
    v2f Bv;
    Bv[0] = Breg[2 * q + 0];
    Bv[1] = Breg[2 * q + 1];
    acc = __builtin_amdgcn_wmma_f32_16x16x4_f32(
        /*neg_a=*/false, A, /*neg_b=*/false, Bv,
        /*c_mod=*/(short)0, acc, /*reuse_a=*/false, /*reuse_b=*/false);
  }

  // Column n==0 -> l3=0 block at [L*64 + u]; columns 1..5 -> l3=2 block.
  const unsigned offA = rowBaseOff + (unsigned)(L * 64 + u0);
  const unsigned offB = rowBaseOff +
      (unsigned)(L3OFF + ((L - 1) * 64 + u0) * 5 + (n - 1));
  unsigned off = (n == 0) ? offA : offB;
  const unsigned stride = (n == 0) ? 1u : 5u;   // elements per row step
  if (n <= 5) {
#pragma unroll
    for (int r = 0; r < 8; ++r) {
      __builtin_nontemporal_store(acc[r], out + off);
      off += stride;
    }
  }
}

__global__ __launch_bounds__(256) void cgtp_wmma_kernel(
    const float* __restrict__ x1, const float* __restrict__ x2,
    const float* __restrict__ cg00, const float* __restrict__ cg10,
    const float* __restrict__ cg20, const float* __restrict__ cg30,
    const float* __restrict__ cg12, const float* __restrict__ cg22,
    const float* __restrict__ cg32,
    float* __restrict__ out, int ntiles)
{
  const int  lane = threadIdx.x & 31;
  const int  n    = lane & 15;
  const bool hi   = lane >= 16;
  const int  wid    = (int)((blockIdx.x * blockDim.x + threadIdx.x) >> 5);
  const int  nwaves = (int)((gridDim.x * blockDim.x) >> 5);

  // CG coefficients held in registers for the whole grid-stride loop.
  float B1[6], B2[14], B3[26];
  load_B<1, 3 >(cg10, cg12, hi, n, B1);   // K=9  -> 3 chunks
  load_B<2, 7 >(cg20, cg22, hi, n, B2);   // K=25 -> 7 chunks
  load_B<3, 13>(cg30, cg32, hi, n, B3);   // K=49 -> 13 chunks
  const float c00 = cg00[0];

  for (int tile = wid; tile < ntiles; tile += nwaves) {
    const int row0 = tile << 4;
    const int row  = row0 + n;            // this lane's (b,u) row
    float r1[16], r2[16];
    const float4* p1 = (const float4*)(x1 + (size_t)row * 16);
    const float4* p2 = (const float4*)(x2 + (size_t)row * 16);
#pragma unroll
    for (int c = 0; c < 4; ++c) {
      float4 a4 = p1[c], b4 = p2[c];
      r1[4 * c + 0] = a4.x; r1[4 * c + 1] = a4.y;
      r1[4 * c + 2] = a4.z; r1[4 * c + 3] = a4.w;
      r2[4 * c + 0] = b4.x; r2[4 * c + 1] = b4.y;
      r2[4 * c + 2] = b4.z; r2[4 * c + 3] = b4.w;
    }

    // bb constant across the tile; u0 in {0,8,16,...,56}.
    const unsigned rowBaseOff = (unsigned)(row0 >> 6) * NCOL;
    const int u0 = (row0 & 63) + (hi ? 8 : 0);

    // l=0, l3=0 path: single scalar product per row (low half stores).
    if (!hi) {
      __builtin_nontemporal_store(r1[0] * r2[0] * c00,
                                  out + rowBaseOff + (unsigned)(row & 63));
    }

    do_path<1, 3 >(r1, r2, B1, hi, n, rowBaseOff, u0, out);
    do_path<2, 7 >(r1, r2, B2, hi, n, rowBaseOff, u0, out);
    do_path<3, 13>(r1, r2, B3, hi, n, rowBaseOff, u0, out);
  }
}

extern "C" void kernel_launch(void* const* d_in, const int* in_sizes, int n_in,
                              void* d_out, int out_size, void* d_ws, size_t ws_size,
                              hipStream_t stream) {
  const float* x1   = (const float*)d_in[0];
  const float* x2   = (const float*)d_in[1];
  const float* cg00 = (const float*)d_in[2];
  const float* cg10 = (const float*)d_in[3];
  const float* cg20 = (const float*)d_in[4];
  const float* cg30 = (const float*)d_in[5];
  const float* cg12 = (const float*)d_in[6];
  const float* cg22 = (const float*)d_in[7];
  const float* cg32 = (const float*)d_in[8];
  float* out = (float*)d_out;

  const int rows   = in_sizes[0] / 16;  // BATCH*MUL = 4,194,304
  const int ntiles = rows / 16;         // 262,144 (exact, no tail)

  dim3 block(256);   // 8 wave32 per block
  dim3 grid(2048);   // grid-stride: ~16 tiles per wave, amortizes CG preload
  hipLaunchKernelGGL(cgtp_wmma_kernel, grid, block, 0, stream,
                     x1, x2, cg00, cg10, cg20, cg30, cg12, cg22, cg32,
                     out, ntiles);
}